// LinearSelfAttentionTemporal_83803401880452
// MI455X (gfx1250) — compile-verified
//
#include <hip/hip_runtime.h>
#include <math.h>

// ---------------------------------------------------------------------------
// Problem constants (from reference):  x:(64,128,64,25)
// ---------------------------------------------------------------------------
#define NB   64              // batch
#define CC   128             // channels
#define TT   64
#define VV   25
#define LL   (TT * VV)       // 1600, sequence length per (n,h)
#define HH   8               // heads
#define HD   16              // head dim  (CC / HH)
#define LTILES (LL / 16)     // 100 l-tiles of 16
#define WSTRIDE 132          // padded LDS stride for 128-wide weight rows
                             // (o*132 mod 64 = 4*o mod 64 -> conflict-free b64,
                             //  528B row pitch -> 16B aligned for b128 staging)

typedef float v2f __attribute__((ext_vector_type(2)));
typedef float v8f __attribute__((ext_vector_type(8)));

// D(16l x 16o) = A(16l x 4c) * B(4c x 16o) + C   accumulated over c
static __device__ __forceinline__ v8f wmma_f32(v2f a, v2f b, v8f c) {
    return __builtin_amdgcn_wmma_f32_16x16x4_f32(
        /*neg_a=*/false, a, /*neg_b=*/false, b,
        /*c_mod=*/(short)0, c, /*reuse_a=*/false, /*reuse_b=*/false);
}

// ---------------------------------------------------------------------------
// Kernel 1:  w[n,h,l,d] = sum_c x[n,c,l] * Wa[h*16+d, c] + ba[h*16+d]
//   grid (25, 64), block 128 (4 waves); wave -> one l-tile, all 8 o-tiles.
//   Wa staged in LDS (padded), A fragments streamed from global (coalesced),
//   output stored in (N,H,L,16) layout: lane-contiguous d -> 64B stores.
// ---------------------------------------------------------------------------
__global__ void lsat_gemm1(const float* __restrict__ x,
                           const float* __restrict__ Wa,
                           const float* __restrict__ ba,
                           float* __restrict__ wbuf) {
    extern __shared__ float smem[];
    float* sW = smem;                    // 128 rows * WSTRIDE
    float* sB = smem + CC * WSTRIDE;     // 128 bias

    const int tid = threadIdx.x;
    // cooperative stage of Wa (16384 floats) via float4
    for (int i = tid; i < (CC * CC) / 4; i += 128) {
        float4 v = ((const float4*)Wa)[i];
        int o  = i >> 5;                 // /32 float4 per row
        int c4 = (i & 31) << 2;
        *(float4*)&sW[o * WSTRIDE + c4] = v;
    }
    if (tid < CC) sB[tid] = ba[tid];
    __syncthreads();

    const int lane = tid & 31;
    const int half = lane >> 4;          // K-pair selector (ISA A/B layout)
    const int m16  = lane & 15;
    const int n    = blockIdx.y;
    const int ltile = blockIdx.x * 4 + (tid >> 5);
    const int l0   = ltile * 16;

    const float* xn = x + (size_t)n * CC * LL;

    v8f acc[HH];
    #pragma unroll
    for (int ot = 0; ot < HH; ++ot) acc[ot] = (v8f)0.0f;

    for (int k0 = 0; k0 < CC; k0 += 4) {
        const int c = k0 + 2 * half;
        // A fragment: rows l0+m16, channels c / c+1 (stride LL floats apart)
        const float* xc = xn + (size_t)c * LL + (l0 + m16);
        v2f a;
        a.x = xc[0];
        a.y = xc[LL];
        #pragma unroll
        for (int ot = 0; ot < HH; ++ot) {
            // B fragment: Wa^T(c..c+1, o) = contiguous pair in LDS row o
            const int o = ot * 16 + m16;
            v2f b = *(const v2f*)&sW[o * WSTRIDE + c];
            acc[ot] = wmma_f32(a, b, acc[ot]);
        }
    }

    // epilogue: D row M=r+8*half -> l, column N=m16 -> d; add bias
    #pragma unroll
    for (int ot = 0; ot < HH; ++ot) {
        const float bias = sB[ot * 16 + m16];
        float* wp = wbuf + ((((size_t)n * HH + ot) * LL) + l0) * HD + m16;
        #pragma unroll
        for (int r = 0; r < 8; ++r) {
            const int lrow = r + 8 * half;
            wp[(size_t)lrow * HD] = acc[ot][r] + bias;
        }
    }
}

// ---------------------------------------------------------------------------
// Kernel 2: per-(n,h) sequential scan along L (two passes, online softmax).
//   One wave per chain; lanes 0..15 carry d (16..31 mirror for full EXEC).
//   Pass 1: cumsum(w^2), logit = temp[h]*sum_d(w^2/denom + dbias[h]),
//           online (m,s) softmax state, logits spilled to tmpbuf.
//   Pass 2: Pi, cumsum(Pi), cumsum(w^2*Pi), attn, y -> wbuf IN PLACE.
// ---------------------------------------------------------------------------
__global__ void lsat_scan(float* __restrict__ wbuf,
                          float* __restrict__ tmpbuf,
                          const float* __restrict__ temp,
                          const float* __restrict__ denom_bias) {
    const int lane  = threadIdx.x & 31;
    const int d     = lane & 15;
    const int chain = blockIdx.x * 4 + (threadIdx.x >> 5);   // n*8 + h
    const int h     = chain & (HH - 1);

    float* wc = wbuf + (size_t)chain * LL * HD;
    float* tc = tmpbuf + (size_t)chain * LL;
    const float tempH = temp[h];
    const float dbias = denom_bias[h];

    // ---- pass 1 ----
    float denomRun = 0.0f;
    float m = -INFINITY, s = 0.0f;
    for (int l = 0; l < LL; ++l) {
        const float wv  = wc[(size_t)l * HD + d];
        const float wsq = wv * wv;
        denomRun += wsq;
        float t = wsq / fmaxf(denomRun, 1e-12f) + dbias;
        // butterfly sum over the 16-lane d group
        t += __shfl_xor(t, 1, 32);
        t += __shfl_xor(t, 2, 32);
        t += __shfl_xor(t, 4, 32);
        t += __shfl_xor(t, 8, 32);
        t *= tempH;
        if (lane == 0) tc[l] = t;
        const float mn = fmaxf(m, t);
        s = s * __expf(m - mn) + __expf(t - mn);
        m = mn;
    }

    // ---- pass 2 ----
    float cumPi = 0.0f, cumWP = 0.0f;
    for (int l = 0; l < LL; ++l) {
        const float t  = tc[l];
        const float Pi = __expf(t - m) / s;
        cumPi += Pi;
        const float wv  = wc[(size_t)l * HD + d];
        const float wsq = wv * wv;
        cumWP += wsq * Pi;
        const float attn = 1.0f / (1.0f + cumWP / (cumPi + 1e-8f));
        const float y = -(wv * Pi) * attn;
        if (lane < 16) wc[(size_t)l * HD + d] = y;   // in-place, coalesced 64B
    }
}

// ---------------------------------------------------------------------------
// Kernel 3: out[n,o,l] = relu( sum_c Wp[o,c]*y[n,c,l] + bp[o] + x[n,o,l] )
//   Roles swapped vs gemm1: A = Wp rows (LDS), B = y (global) so the
//   epilogue stores / residual loads are l-contiguous and coalesced.
//   y is read from the (N,H,L,16) layout (c -> (h=c>>4, d=c&15)); the two
//   K-values per lane are d,d+1 -> single 8-byte load.
// ---------------------------------------------------------------------------
__global__ void lsat_gemm2(const float* __restrict__ ybuf,
                           const float* __restrict__ Wp,
                           const float* __restrict__ bp,
                           const float* __restrict__ x,
                           float* __restrict__ out) {
    extern __shared__ float smem[];
    float* sW = smem;
    float* sB = smem + CC * WSTRIDE;

    const int tid = threadIdx.x;
    for (int i = tid; i < (CC * CC) / 4; i += 128) {
        float4 v = ((const float4*)Wp)[i];
        int o  = i >> 5;
        int c4 = (i & 31) << 2;
        *(float4*)&sW[o * WSTRIDE + c4] = v;
    }
    if (tid < CC) sB[tid] = bp[tid];
    __syncthreads();

    const int lane = tid & 31;
    const int half = lane >> 4;
    const int m16  = lane & 15;
    const int n    = blockIdx.y;
    const int ltile = blockIdx.x * 4 + (tid >> 5);
    const int l0   = ltile * 16;

    v8f acc[HH];
    #pragma unroll
    for (int ot = 0; ot < HH; ++ot) acc[ot] = (v8f)0.0f;

    for (int k0 = 0; k0 < CC; k0 += 4) {
        const int c = k0 + 2 * half;       // even -> d, d+1 stay in one head
        const int hh = c >> 4;
        const int dd = c & 15;
        // B fragment: y[n, hh, l0+m16, dd..dd+1]
        const float* yp = ybuf + ((((size_t)n * HH + hh) * LL) + (l0 + m16)) * HD + dd;
        v2f b = *(const v2f*)yp;
        #pragma unroll
        for (int ot = 0; ot < HH; ++ot) {
            // A fragment: Wp rows o = ot*16+m16, channels c..c+1 from LDS
            const int o = ot * 16 + m16;
            v2f a = *(const v2f*)&sW[o * WSTRIDE + c];
            acc[ot] = wmma_f32(a, b, acc[ot]);
        }
    }

    // epilogue: D row M -> o, column N=m16 -> l; bias + residual + relu
    const float* xn = x + (size_t)n * CC * LL;
    float* on = out + (size_t)n * CC * LL;
    #pragma unroll
    for (int ot = 0; ot < HH; ++ot) {
        #pragma unroll
        for (int r = 0; r < 8; ++r) {
            const int o = ot * 16 + r + 8 * half;
            const size_t idx = (size_t)o * LL + (l0 + m16);
            const float v = acc[ot][r] + sB[o] + xn[idx];
            on[idx] = fmaxf(v, 0.0f);
        }
    }
}

// ---------------------------------------------------------------------------
// Launch
// ---------------------------------------------------------------------------
extern "C" void kernel_launch(void* const* d_in, const int* in_sizes, int n_in,
                              void* d_out, int out_size, void* d_ws, size_t ws_size,
                              hipStream_t stream) {
    const float* x    = (const float*)d_in[0];
    const float* Wa   = (const float*)d_in[1];
    const float* ba   = (const float*)d_in[2];
    const float* Wp   = (const float*)d_in[3];
    const float* bp   = (const float*)d_in[4];
    const float* temp = (const float*)d_in[5];
    const float* dbia = (const float*)d_in[6];
    float* out = (float*)d_out;

    // workspace: w/y buffer (N*H*L*16 = 13.1M floats) + logits (N*H*L)
    float* wbuf   = (float*)d_ws;
    float* tmpbuf = wbuf + (size_t)NB * HH * LL * HD;

    const size_t shmem = (size_t)(CC * WSTRIDE + CC) * sizeof(float); // ~68KB

    dim3 ggrid(LTILES / 4, NB);   // (25, 64), 4 waves/block -> 1 l-tile each
    lsat_gemm1<<<ggrid, 128, shmem, stream>>>(x, Wa, ba, wbuf);
    lsat_scan <<<(NB * HH) / 4, 128, 0, stream>>>(wbuf, tmpbuf, temp, dbia);
    lsat_gemm2<<<ggrid, 128, shmem, stream>>>(wbuf, Wp, bp, x, out);
}